// NAM_42442866819214
// MI455X (gfx1250) — compile-verified
//
#include <hip/hip_runtime.h>

#define B_SZ  8192
#define F_SZ  128
#define H1_SZ 64
#define H2_SZ 32
#define BT    64        // batch rows per block

typedef __attribute__((ext_vector_type(2))) float v2f;
typedef __attribute__((ext_vector_type(8))) float v8f;
typedef __attribute__((ext_vector_type(4))) unsigned int v4u;
typedef __attribute__((ext_vector_type(8))) int v8i;
typedef __attribute__((ext_vector_type(4))) int v4i;

// One block = one feature f, 64 batch rows. 4 wave32s:
//   wave = {rowPair 0..1} x {colTile 0..1}; each wave owns 32 rows x 16 cols
//   as TWO 16x16 WMMA accumulators so every B fragment feeds 2 WMMAs.
// W2 (8KB) is staged global->LDS by the Tensor Data Mover (1D tile descriptor),
// synchronized with s_wait_tensorcnt + workgroup barrier.
__global__ __launch_bounds__(128) void nam_feature_mlp(
    const float* __restrict__ inputs,   // [B,F]
    const float* __restrict__ W1,       // [F,H1]
    const float* __restrict__ b1,       // [F,H1]
    const float* __restrict__ W2,       // [F,H1,H2]
    const float* __restrict__ b2,       // [F,H2]
    const float* __restrict__ W3,       // [F,H2]
    const float* __restrict__ b3,       // [F]
    float* __restrict__ contribs)       // [B,F]  (dropout_outputs)
{
    __shared__ float sW2[H1_SZ * H2_SZ];   // 8 KB, TDM destination
    __shared__ float sW1[H1_SZ];
    __shared__ float sB1[H1_SZ];
    __shared__ float sB2[H2_SZ];
    __shared__ float sW3[H2_SZ];
    __shared__ float sPart[BT * 2];        // [row in tile][colTile]

    const int f    = blockIdx.y;
    const int tid  = threadIdx.x;
    const int wave = tid >> 5;

    // ---- TDM: DMA this feature's W2 block (2048 f32, contiguous) into LDS ----
    if (wave == 0) {
        const unsigned long long gaddr =
            (unsigned long long)(const void*)(W2 + (size_t)f * (H1_SZ * H2_SZ));
        const unsigned int ldsaddr = (unsigned int)(size_t)(void*)sW2; // LDS aperture low 32b

        v4u g0;
        g0[0] = 1u;                                            // count=1, user desc
        g0[1] = ldsaddr;                                       // lds_addr
        g0[2] = (unsigned int)(gaddr & 0xFFFFFFFFu);           // global_addr[31:0]
        g0[3] = (unsigned int)((gaddr >> 32) & 0x1FFFFFFu)     // global_addr[56:32]
              | (2u << 30);                                    // type=2 ("image")

        v8i g1;
        g1[0] = (int)(2u << 16);        // workgroup_mask=0 (no cluster), data_size=4B
        g1[1] = (int)(2048u << 16);     // tensor_dim0[15:0]=2048 @ [63:48]
        g1[2] = (int)(1u << 16);        // tensor_dim0 hi=0; tensor_dim1[15:0]=1 @ [95:80]
        g1[3] = (int)(2048u << 16);     // tile_dim0=2048 @ [127:112]
        g1[4] = 1;                      // tile_dim1=1 (tile_dim2=0 unused)
        g1[5] = 2048;                   // tensor_dim0_stride (low 32)
        g1[6] = 0;
        g1[7] = 0;

        v4i gz4 = {0, 0, 0, 0};                      // groups 2/3 unused (<=2D tensor)
        v8i gz8 = {0, 0, 0, 0, 0, 0, 0, 0};
        __builtin_amdgcn_tensor_load_to_lds(g0, g1, gz4, gz4, gz8, /*cpol=*/0);
    }

    // ---- small parameter vectors via plain b128 loads ----
    if (tid < 16)
        ((float4*)sW1)[tid]      = ((const float4*)(W1 + (size_t)f * H1_SZ))[tid];
    else if (tid < 32)
        ((float4*)sB1)[tid - 16] = ((const float4*)(b1 + (size_t)f * H1_SZ))[tid - 16];
    else if (tid < 40)
        ((float4*)sB2)[tid - 32] = ((const float4*)(b2 + (size_t)f * H2_SZ))[tid - 32];
    else if (tid < 48)
        ((float4*)sW3)[tid - 40] = ((const float4*)(W3 + (size_t)f * H2_SZ))[tid - 40];

    if (wave == 0)
        __builtin_amdgcn_s_wait_tensorcnt(0);   // W2 tile resident before publishing
    __syncthreads();

    const int lane    = tid & 31;
    const int colTile = wave & 1;
    const int rowPair = wave >> 1;
    const int l16     = lane & 15;
    const int kh      = lane >> 4;   // lane half selects K pairs (0,1) vs (2,3)

    const int rbase = blockIdx.x * BT + rowPair * 32;
    const int row0  = rbase + l16;          // rows for accumulator 0
    const int row1  = rbase + 16 + l16;     // rows for accumulator 1
    const int col   = colTile * 16 + l16;   // N index = h2 unit

    const float x0  = inputs[(size_t)row0 * F_SZ + f];
    const float x1  = inputs[(size_t)row1 * F_SZ + f];
    const float w3c = sW3[col];
    const float b2c = sB2[col];

    v8f acc0 = {};
    v8f acc1 = {};
    #pragma unroll
    for (int s = 0; s < 16; ++s) {
        const int ka = s * 4 + kh * 2;  // A vgpr0: K=ka, vgpr1: K=ka+1 (ISA layout)
        const float w1a = sW1[ka], w1b = sW1[ka + 1];
        const float c1a = sB1[ka], c1b = sB1[ka + 1];
        v2f b;
        b.x = sW2[ka * H2_SZ + col];
        b.y = sW2[(ka + 1) * H2_SZ + col];

        float p0 = fmaf(x0, w1a, c1a), p1 = fmaf(x0, w1b, c1b);
        float q0 = fmaf(x1, w1a, c1a), q1 = fmaf(x1, w1b, c1b);
        v2f a0, a1;
        a0.x = p0 > 0.0f ? p0 : 0.0f;  a0.y = p1 > 0.0f ? p1 : 0.0f;
        a1.x = q0 > 0.0f ? q0 : 0.0f;  a1.y = q1 > 0.0f ? q1 : 0.0f;

        acc0 = __builtin_amdgcn_wmma_f32_16x16x4_f32(false, a0, false, b,
                                                     (short)0, acc0, false, false);
        acc1 = __builtin_amdgcn_wmma_f32_16x16x4_f32(false, a1, false, b,
                                                     (short)0, acc1, false, false);
    }

    // ---- epilogue: relu(acc + b2) * W3, row-sum over this wave's 16 cols ----
    // C/D layout: VGPR v -> M=v (lanes 0-15), M=v+8 (lanes 16-31)
    #pragma unroll
    for (int v = 0; v < 8; ++v) {
        float h0 = acc0[v] + b2c;  h0 = h0 > 0.0f ? h0 : 0.0f;
        float h1 = acc1[v] + b2c;  h1 = h1 > 0.0f ? h1 : 0.0f;
        float s0 = h0 * w3c;
        float s1 = h1 * w3c;
        s0 += __shfl_xor(s0, 1);  s1 += __shfl_xor(s1, 1);
        s0 += __shfl_xor(s0, 2);  s1 += __shfl_xor(s1, 2);
        s0 += __shfl_xor(s0, 4);  s1 += __shfl_xor(s1, 4);
        s0 += __shfl_xor(s0, 8);  s1 += __shfl_xor(s1, 8);  // sums within 16-lane halves
        if (l16 == 0) {
            int r = rowPair * 32 + v + kh * 8;
            sPart[r * 2 + colTile]        = s0;
            sPart[(r + 16) * 2 + colTile] = s1;
        }
    }
    __syncthreads();

    if (tid < BT) {
        float c = sPart[tid * 2 + 0] + sPart[tid * 2 + 1] + b3[f];
        int grow = blockIdx.x * BT + tid;
        contribs[(size_t)grow * F_SZ + f] = c;
    }
}

// outputs[b] = bias + sum_f contribs[b,f]  (deterministic tree reduction, no atomics)
__global__ __launch_bounds__(256) void nam_reduce(
    const float* __restrict__ contribs,  // [B,F]
    const float* __restrict__ bias,      // [1]
    float* __restrict__ outputs)         // [B]
{
    const int lane = threadIdx.x & 31;
    const int wave = threadIdx.x >> 5;
    const int row  = blockIdx.x * 8 + wave;

    const float* p = contribs + (size_t)row * F_SZ;
    float s = 0.0f;
    #pragma unroll
    for (int j = 0; j < 4; ++j)
        s += p[lane + 32 * j];
    s += __shfl_xor(s, 1);
    s += __shfl_xor(s, 2);
    s += __shfl_xor(s, 4);
    s += __shfl_xor(s, 8);
    s += __shfl_xor(s, 16);
    if (lane == 0)
        outputs[row] = s + bias[0];
}

extern "C" void kernel_launch(void* const* d_in, const int* in_sizes, int n_in,
                              void* d_out, int out_size, void* d_ws, size_t ws_size,
                              hipStream_t stream) {
    (void)in_sizes; (void)n_in; (void)out_size; (void)d_ws; (void)ws_size;
    const float* inputs = (const float*)d_in[0];
    const float* W1     = (const float*)d_in[1];
    const float* b1     = (const float*)d_in[2];
    const float* W2     = (const float*)d_in[3];
    const float* b2     = (const float*)d_in[4];
    const float* W3     = (const float*)d_in[5];
    const float* b3     = (const float*)d_in[6];
    const float* bias   = (const float*)d_in[7];

    float* outputs  = (float*)d_out;          // [B]
    float* contribs = (float*)d_out + B_SZ;   // [B,F] (dropout_outputs)

    dim3 grid(B_SZ / BT, F_SZ);
    nam_feature_mlp<<<grid, 128, 0, stream>>>(inputs, W1, b1, W2, b2, W3, b3, contribs);
    nam_reduce<<<B_SZ / 8, 256, 0, stream>>>(contribs, bias, outputs);
}